// TimeAttention_38345468019460
// MI455X (gfx1250) — compile-verified
//
#include <hip/hip_runtime.h>
#include <hip/hip_bf16.h>
#include <math.h>

// Problem constants (match reference)
static constexpr int kB    = 64;
static constexpr int kT    = 512;
static constexpr int kC    = 512;
static constexpr int kHALF = 50;
static constexpr int kW    = 2 * kHALF + 1;   // 101
static constexpr int kWPad = 112;             // padded LDS row (16B aligned)
static constexpr int WAVES = 8;               // waves (rows) per block
static constexpr int BLOCK = 32 * WAVES;      // 256 threads

#define GLOBAL_AS __attribute__((address_space(1)))
#define LDS_AS    __attribute__((address_space(3)))

// 16-byte int vector type matching the async builtin's parameter type
typedef int ta_v4i __attribute__((vector_size(4 * sizeof(int))));

// ---- gfx1250 async global->LDS path (guarded; falls back to plain loads) ----
#if defined(__has_builtin)
#  if __has_builtin(__builtin_amdgcn_global_load_async_to_lds_b128)
#    define TA_USE_ASYNC 1
#  endif
#endif

__device__ __forceinline__ void ta_wait_async0() {
#if defined(__has_builtin) && __has_builtin(__builtin_amdgcn_s_wait_asynccnt)
  __builtin_amdgcn_s_wait_asynccnt(0);
#else
  asm volatile("s_wait_asynccnt 0x0" ::: "memory");
#endif
}

__device__ __forceinline__ float wave_reduce_max(float v) {
#pragma unroll
  for (int off = 16; off >= 1; off >>= 1)
    v = fmaxf(v, __shfl_xor(v, off, 32));
  return v;
}

__device__ __forceinline__ float wave_reduce_sum(float v) {
#pragma unroll
  for (int off = 16; off >= 1; off >>= 1)
    v += __shfl_xor(v, off, 32);
  return v;
}

__global__ __launch_bounds__(BLOCK, 4) void ta_softmax_kernel(
    const int* __restrict__ concepts,     // [B,T]
    const int* __restrict__ tgt_ts,       // [B,T]
    const int* __restrict__ ctx_ts,       // [B,C]
    const int* __restrict__ tmask,        // [B,C]
    const float* __restrict__ emb,        // [VOCAB,W]
    float* __restrict__ out)              // [B,T,C]
{
  __shared__ alignas(16) int   s_ctx[kC];
  __shared__ alignas(16) int   s_msk[kC];
  __shared__ alignas(16) float s_emb[WAVES][kWPad];

  const int b    = blockIdx.y;
  const int tid  = threadIdx.x;
  const int wave = tid >> 5;
  const int lane = tid & 31;

  // ---- stage context timestamps + mask for this b (4 KB) ----
#if defined(TA_USE_ASYNC)
  {
    const int t4 = tid * 4;
    if (tid < kC / 4) {
      __builtin_amdgcn_global_load_async_to_lds_b128(
          (GLOBAL_AS ta_v4i*)(ctx_ts + (size_t)b * kC + t4),
          (LDS_AS ta_v4i*)(s_ctx + t4), 0, 0);
    } else {
      const int u = t4 - kC;
      __builtin_amdgcn_global_load_async_to_lds_b128(
          (GLOBAL_AS ta_v4i*)(tmask + (size_t)b * kC + u),
          (LDS_AS ta_v4i*)(s_msk + u), 0, 0);
    }
    ta_wait_async0();
  }
#else
  {
    const int base = b * kC;
    s_ctx[tid]         = ctx_ts[base + tid];
    s_ctx[tid + BLOCK] = ctx_ts[base + tid + BLOCK];
    s_msk[tid]         = tmask[base + tid];
    s_msk[tid + BLOCK] = tmask[base + tid + BLOCK];
  }
#endif

  // ---- per-wave row setup + stage 101-float emb window row ----
  const int row_t   = blockIdx.x * WAVES + wave;     // t index
  const int row     = b * kT + row_t;                // flat (b,t)
  const int concept = concepts[row];                 // uniform within wave
  const int tgt     = tgt_ts[row];                   // uniform within wave

  {
    const float* erow = emb + (size_t)concept * kW;
#pragma unroll
    for (int k = 0; k < 4; ++k) {
      const int i = k * 32 + lane;
      if (i < kW) s_emb[wave][i] = erow[i];
    }
  }
  __syncthreads();   // ctx/mask + all emb rows visible

  // ---- logits: gather from LDS window row, apply mask ----
  float vals[16];
  float vmax = -INFINITY;
#pragma unroll
  for (int k = 0; k < 4; ++k) {
    const int c = k * 128 + lane * 4;
    const int4 cv = *(const int4*)(s_ctx + c);
    const int4 mv = *(const int4*)(s_msk + c);

    int d0 = min(max(cv.x - tgt, -kHALF), kHALF) + kHALF;
    int d1 = min(max(cv.y - tgt, -kHALF), kHALF) + kHALF;
    int d2 = min(max(cv.z - tgt, -kHALF), kHALF) + kHALF;
    int d3 = min(max(cv.w - tgt, -kHALF), kHALF) + kHALF;

    float l0 = s_emb[wave][d0] + (mv.x ? -1e9f : 0.0f);
    float l1 = s_emb[wave][d1] + (mv.y ? -1e9f : 0.0f);
    float l2 = s_emb[wave][d2] + (mv.z ? -1e9f : 0.0f);
    float l3 = s_emb[wave][d3] + (mv.w ? -1e9f : 0.0f);

    vals[k * 4 + 0] = l0;  vals[k * 4 + 1] = l1;
    vals[k * 4 + 2] = l2;  vals[k * 4 + 3] = l3;
    vmax = fmaxf(vmax, fmaxf(fmaxf(l0, l1), fmaxf(l2, l3)));
  }

  // ---- softmax over C (intra-wave only) ----
  vmax = wave_reduce_max(vmax);

  float sum = 0.0f;
#pragma unroll
  for (int i = 0; i < 16; ++i) {
    vals[i] = __expf(vals[i] - vmax);
    sum += vals[i];
  }
  sum = wave_reduce_sum(sum);
  const float inv = 1.0f / sum;

  // ---- coalesced float4 stores ----
  float* orow = out + (size_t)row * kC;
#pragma unroll
  for (int k = 0; k < 4; ++k) {
    const int c = k * 128 + lane * 4;
    float4 o;
    o.x = vals[k * 4 + 0] * inv;
    o.y = vals[k * 4 + 1] * inv;
    o.z = vals[k * 4 + 2] * inv;
    o.w = vals[k * 4 + 3] * inv;
    *(float4*)(orow + c) = o;
  }
}

extern "C" void kernel_launch(void* const* d_in, const int* in_sizes, int n_in,
                              void* d_out, int out_size, void* d_ws, size_t ws_size,
                              hipStream_t stream) {
  const int*   concepts = (const int*)d_in[0];   // target_concepts  [B,T]
  const int*   tgt_ts   = (const int*)d_in[1];   // target_time_stamps [B,T]
  const int*   ctx_ts   = (const int*)d_in[2];   // context_time_stamps [B,C]
  const int*   tmask    = (const int*)d_in[3];   // time_mask [B,C]
  const float* emb      = (const float*)d_in[4]; // time_emb [VOCAB,W]
  float*       out      = (float*)d_out;         // [B,T,C]

  dim3 grid(kT / WAVES, kB);   // 64 x 64 blocks, 8 rows (waves) per block
  ta_softmax_kernel<<<grid, BLOCK, 0, stream>>>(concepts, tgt_ts, ctx_ts,
                                                tmask, emb, out);
}